// Digit_Caps_8564164788868
// MI455X (gfx1250) — compile-verified
//
#include <hip/hip_runtime.h>
#include <math.h>

// ---------------------------------------------------------------------------
// DigitCaps dynamic routing, fused & recompute-based, for MI455X (gfx1250).
//
// u_hat (1 GiB fp32) is never materialized. W (128 MiB) + x (16 MiB) fit in
// the 192 MB L2, so three recompute sweeps stream from L2 instead of HBM.
// Routing algebra: logits after step k == u_hat . (v1 + ... + vk), and the
// first coupling is uniform 1/32, so no logit storage is needed at all.
// Core math uses V_WMMA_F32_16X16X4_F32 (exact fp32 tensor-core path).
// ---------------------------------------------------------------------------

typedef __attribute__((ext_vector_type(2))) float v2f;
typedef __attribute__((ext_vector_type(8))) float v8f;

#define BATCH 128
#define J_CAP 2048
#define I_DIM 16
#define O_CAP 32
#define D_DIM 32

#define BT 16              // batch rows per workgroup (WMMA M)
#define JC 32              // input capsules per workgroup chunk
#define NWAVES 16          // one wave per 2 output capsules
#define BLOCK (NWAVES * 32)
#define XROW (JC * I_DIM + 1)   // padded LDS row stride (513 -> conflict-free)

static_assert(JC * I_DIM == 512, "x staging indexing assumes 512");

// MODE 0: first sweep, coupling c == 1/32 (softmax of zero logits).
// MODE 1: coupling c = softmax_o( u_hat . v ), v passed in.
template <int MODE>
__global__ __launch_bounds__(BLOCK) void caps_sweep(
    const float* __restrict__ x,   // [B, J, I]
    const float* __restrict__ W,   // [O, J, D, I]
    const float* __restrict__ v,   // [B, O, D]  (MODE 1 only)
    float* __restrict__ s_out)     // [B, O, D]  accumulated via atomics
{
    __shared__ float xs[BT * XROW];
    __shared__ float agreeLDS[BT][O_CAP + 1];
    __shared__ float cLDS[BT][O_CAP + 1];

    const int tid    = threadIdx.x;
    const int wave   = tid >> 5;
    const int lane   = tid & 31;
    const int l16    = lane & 15;
    const int hiHalf = lane >> 4;          // 0: lanes 0-15, 1: lanes 16-31
    const int bbase  = blockIdx.x * BT;
    const int j0     = blockIdx.y * JC;
    const int o0     = wave * 2;

    // ---- stage x tile [BT x JC x I] into LDS (coalesced) ----
    const float* xg = x + (size_t)bbase * J_CAP * I_DIM + (size_t)j0 * I_DIM;
    for (int idx = tid; idx < BT * JC * I_DIM; idx += BLOCK) {
        const int bl  = idx >> 9;          // / 512
        const int rem = idx & 511;
        xs[bl * XROW + rem] = xg[(size_t)bl * J_CAP * I_DIM + rem];
    }
    __syncthreads();

    // ---- preload v in WMMA C-tile layout (MODE 1) ----
    // C tile row r, lanes 0-15 -> (b=bbase+r, d=h*16+l16); lanes 16-31 -> b+8.
    v8f vt[2][2];
    if (MODE == 1) {
#pragma unroll
        for (int op = 0; op < 2; ++op)
#pragma unroll
            for (int h = 0; h < 2; ++h)
#pragma unroll
                for (int r = 0; r < 8; ++r) {
                    const int b = bbase + r + hiHalf * 8;
                    vt[op][h][r] =
                        v[((size_t)b * O_CAP + (o0 + op)) * D_DIM + h * 16 + l16];
                }
    }

    v8f s_acc[2][2];
#pragma unroll
    for (int op = 0; op < 2; ++op)
#pragma unroll
        for (int h = 0; h < 2; ++h)
            s_acc[op][h] = (v8f){0, 0, 0, 0, 0, 0, 0, 0};

    for (int jj = 0; jj < JC; ++jj) {
        const int j = j0 + jj;

        // A fragments: x tile rows, K=16 split into 4 wmma steps.
        // f32 A layout: VGPR0 = K0 (lanes 0-15) | K2 (lanes 16-31); VGPR1 = K1 | K3.
        v2f a[4];
        const float* xrow = xs + l16 * XROW + jj * I_DIM;
#pragma unroll
        for (int kk = 0; kk < 4; ++kk) {
            a[kk].x = xrow[4 * kk + hiHalf * 2 + 0];
            a[kk].y = xrow[4 * kk + hiHalf * 2 + 1];
        }

        // u_hat tiles: C[op][h] covers (16 b) x (d = h*16 .. h*16+15) for o0+op.
        // B fragment layout mirrors A/C: VGPR0 = K0|K2, VGPR1 = K1|K3, so each
        // half-wave needs the contiguous pair W[o,j,d, 4kk + 2*hiHalf + {0,1}]:
        // one 8-byte load per lane, no selects, no redundant bytes.
        v8f C[2][2];
#pragma unroll
        for (int op = 0; op < 2; ++op) {
            const size_t wbase = ((size_t)(o0 + op) * J_CAP + j) * (size_t)(D_DIM * I_DIM);
#pragma unroll
            for (int h = 0; h < 2; ++h) {
                v8f c = (v8f){0, 0, 0, 0, 0, 0, 0, 0};
                const float* wrow = W + wbase + (size_t)(h * 16 + l16) * I_DIM + hiHalf * 2;
#pragma unroll
                for (int kk = 0; kk < 4; ++kk) {
                    const float2 wv = *(const float2*)(wrow + 4 * kk);
                    v2f bfrag;
                    bfrag.x = wv.x;   // K = 4kk + (hiHalf ? 2 : 0)
                    bfrag.y = wv.y;   // K = 4kk + (hiHalf ? 3 : 1)
                    c = __builtin_amdgcn_wmma_f32_16x16x4_f32(
                        false, a[kk], false, bfrag, (short)0, c, false, false);
                }
                C[op][h] = c;
            }
        }

        v8f ct[2];
        if (MODE == 1) {
            // agree[b,o] = sum_d u_hat[b,o,j,d] * v[b,o,d]
            // reduce over the 16 d-lanes within each half wave.
#pragma unroll
            for (int op = 0; op < 2; ++op) {
                v8f p = C[op][0] * vt[op][0] + C[op][1] * vt[op][1];
#pragma unroll
                for (int r = 0; r < 8; ++r) {
                    float pr = p[r];
                    pr += __shfl_xor(pr, 1, 32);
                    pr += __shfl_xor(pr, 2, 32);
                    pr += __shfl_xor(pr, 4, 32);
                    pr += __shfl_xor(pr, 8, 32);
                    if (l16 == 0)
                        agreeLDS[r + hiHalf * 8][o0 + op] = pr;
                }
            }
            __syncthreads();

            // softmax over o: remap wave <-> batch row, lane <-> o.
            {
                const float aval = agreeLDS[wave][lane];
                float mx = aval;
                mx = fmaxf(mx, __shfl_xor(mx, 16, 32));
                mx = fmaxf(mx, __shfl_xor(mx, 8, 32));
                mx = fmaxf(mx, __shfl_xor(mx, 4, 32));
                mx = fmaxf(mx, __shfl_xor(mx, 2, 32));
                mx = fmaxf(mx, __shfl_xor(mx, 1, 32));
                const float e = __expf(aval - mx);
                float se = e;
                se += __shfl_xor(se, 16, 32);
                se += __shfl_xor(se, 8, 32);
                se += __shfl_xor(se, 4, 32);
                se += __shfl_xor(se, 2, 32);
                se += __shfl_xor(se, 1, 32);
                cLDS[wave][lane] = e / se;
            }
            __syncthreads();

            // re-broadcast coupling in C-tile layout
#pragma unroll
            for (int op = 0; op < 2; ++op)
#pragma unroll
                for (int r = 0; r < 8; ++r)
                    ct[op][r] = cLDS[r + hiHalf * 8][o0 + op];
        }

        // accumulate s += c * u_hat (c == 1/32 folded at flush for MODE 0)
#pragma unroll
        for (int op = 0; op < 2; ++op)
#pragma unroll
            for (int h = 0; h < 2; ++h) {
                if (MODE == 0)
                    s_acc[op][h] = s_acc[op][h] + C[op][h];
                else
                    s_acc[op][h] = s_acc[op][h] + ct[op] * C[op][h];
            }
    }

    // ---- flush partial s to global with fp32 atomics ----
    const float scale = (MODE == 0) ? (1.0f / 32.0f) : 1.0f;
#pragma unroll
    for (int op = 0; op < 2; ++op)
#pragma unroll
        for (int h = 0; h < 2; ++h)
#pragma unroll
            for (int r = 0; r < 8; ++r) {
                const int b = bbase + r + hiHalf * 8;
                atomicAdd(&s_out[((size_t)b * O_CAP + (o0 + op)) * D_DIM + h * 16 + l16],
                          s_acc[op][h][r] * scale);
            }
}

// vout = (vprev ? vprev : 0) + squash(s), one thread per (b, o) row.
__global__ void caps_squash(const float* __restrict__ s,
                            const float* __restrict__ vprev,
                            float* __restrict__ vout)
{
    const int row = blockIdx.x * blockDim.x + threadIdx.x;
    if (row >= BATCH * O_CAP) return;
    const float* sp = s + (size_t)row * D_DIM;
    float tmp[D_DIM];
    float sq = 0.0f;
#pragma unroll
    for (int d = 0; d < D_DIM; ++d) {
        tmp[d] = sp[d];
        sq += tmp[d] * tmp[d];
    }
    const float f = (sq / (1.0f + sq)) / (sqrtf(sq) + 1e-8f);
    float* vp = vout + (size_t)row * D_DIM;
    if (vprev) {
        const float* pp = vprev + (size_t)row * D_DIM;
#pragma unroll
        for (int d = 0; d < D_DIM; ++d) vp[d] = pp[d] + tmp[d] * f;
    } else {
#pragma unroll
        for (int d = 0; d < D_DIM; ++d) vp[d] = tmp[d] * f;
    }
}

extern "C" void kernel_launch(void* const* d_in, const int* in_sizes, int n_in,
                              void* d_out, int out_size, void* d_ws, size_t ws_size,
                              hipStream_t stream)
{
    (void)in_sizes; (void)n_in; (void)out_size; (void)ws_size;
    const float* x = (const float*)d_in[0];   // [128, 2048, 16]
    const float* W = (const float*)d_in[1];   // [1, 32, 2048, 32, 16]
    float* out = (float*)d_out;               // [128, 32, 32]

    const size_t SVN = (size_t)BATCH * O_CAP * D_DIM;  // 131072 floats
    float* s  = (float*)d_ws;     // accumulator          (512 KB)
    float* v1 = s + SVN;          // v after iteration 1  (512 KB)
    float* vs = v1 + SVN;         // v1 + v2              (512 KB)

    dim3 grid(BATCH / BT, J_CAP / JC);   // 8 x 64 workgroups
    dim3 blk(BLOCK);                     // 512 threads = 16 waves
    dim3 qgrid((BATCH * O_CAP + 255) / 256);
    dim3 qblk(256);

    // sweep 1: uniform coupling 1/32 -> s1 ; v1 = squash(s1)
    hipMemsetAsync(s, 0, SVN * sizeof(float), stream);
    caps_sweep<0><<<grid, blk, 0, stream>>>(x, W, nullptr, s);
    caps_squash<<<qgrid, qblk, 0, stream>>>(s, nullptr, v1);

    // sweep 2: logits = u_hat . v1 -> s2 ; vs = v1 + squash(s2)
    hipMemsetAsync(s, 0, SVN * sizeof(float), stream);
    caps_sweep<1><<<grid, blk, 0, stream>>>(x, W, v1, s);
    caps_squash<<<qgrid, qblk, 0, stream>>>(s, v1, vs);

    // sweep 3: logits = u_hat . (v1+v2) -> s3 ; out = squash(s3)
    hipMemsetAsync(s, 0, SVN * sizeof(float), stream);
    caps_sweep<1><<<grid, blk, 0, stream>>>(x, W, vs, s);
    caps_squash<<<qgrid, qblk, 0, stream>>>(s, nullptr, out);
}